// AdaAttN_13391708029240
// MI455X (gfx1250) — compile-verified
//
#include <hip/hip_runtime.h>
#include <hip/hip_bf16.h>

typedef __attribute__((ext_vector_type(2))) float v2f;
typedef __attribute__((ext_vector_type(8))) float v8f;

#define Bc   2
#define Cc   512
#define HWc  2304
#define Mc   2304
#define TOTc 2359296L          /* B*C*HW = B*C*M */
#define ATc  5308416L          /* HW*M */
#define TOLc 4.608e-4f         /* B*HW*1e-7 */

/* ------------------------------------------------------------------ */
/* Generic f32 WMMA GEMM: C[b] = A[b] x B[b] (+bias) (-sub)           */
/* block tile 64x128, BK=32, 8 wave32 each computing 32x32.           */
/* A tile row-major stride 34; B tile TRANSPOSED (col-major, stride   */
/* 34) so both A and B fragments are single contiguous b64 LDS loads. */
/* ------------------------------------------------------------------ */
__launch_bounds__(256)
__global__ void wmma_gemm_f32(
    const float* __restrict__ Ap, long sa_r, long sa_c, long sa_b,
    const float* __restrict__ Bp, long sb_r, long sb_c, long sb_b,
    float* __restrict__ Cp, long sc_r, long sc_b,
    const float* __restrict__ bias, long sbi_r, long sbi_c,
    const float* __restrict__ sub, int K)
{
    __shared__ __align__(16) float As[64 * 34];
    __shared__ __align__(16) float BsT[128 * 34];   /* BsT[n*34 + k] */

    const int b = blockIdx.z;
    Ap += (long)b * sa_b;
    Bp += (long)b * sb_b;
    Cp += (long)b * sc_b;
    if (sub) sub += (long)b * sc_b;

    const int m0 = blockIdx.y * 64;
    const int n0 = blockIdx.x * 128;
    const int tid  = threadIdx.x;
    const int lane = tid & 31;
    const int wave = tid >> 5;
    const int mw = (wave >> 2) * 32;   /* wave row offset in tile */
    const int nw = (wave & 3) * 32;    /* wave col offset in tile */
    const int lh = lane >> 4;          /* 0: k+0..1, 1: k+2..3   */
    const int lm = lane & 15;

    v8f acc00 = {}, acc01 = {}, acc10 = {}, acc11 = {};

    for (int kb = 0; kb < K; kb += 32) {
#pragma unroll
        for (int i = 0; i < 8; ++i) {          /* A tile 64x32 */
            int e = tid + (i << 8);
            int r = e >> 5, c = e & 31;
            As[r * 34 + c] = Ap[(long)(m0 + r) * sa_r + (long)(kb + c) * sa_c];
        }
#pragma unroll
        for (int i = 0; i < 16; ++i) {         /* B tile 32x128, stored transposed */
            int e = tid + (i << 8);
            int r = e >> 7, c = e & 127;       /* r = k, c = n (coalesced on n) */
            BsT[c * 34 + r] = Bp[(long)(kb + r) * sb_r + (long)(n0 + c) * sb_c];
        }
        __syncthreads();

#pragma unroll
        for (int ks = 0; ks < 8; ++ks) {
            const int kk = ks * 4 + lh * 2;
            v2f a0 = *(const v2f*)&As[(mw + lm) * 34 + kk];
            v2f a1 = *(const v2f*)&As[(mw + 16 + lm) * 34 + kk];
            v2f b0 = *(const v2f*)&BsT[(nw + lm) * 34 + kk];
            v2f b1 = *(const v2f*)&BsT[(nw + 16 + lm) * 34 + kk];
            acc00 = __builtin_amdgcn_wmma_f32_16x16x4_f32(false, a0, false, b0, (short)0, acc00, false, false);
            acc01 = __builtin_amdgcn_wmma_f32_16x16x4_f32(false, a0, false, b1, (short)0, acc01, false, false);
            acc10 = __builtin_amdgcn_wmma_f32_16x16x4_f32(false, a1, false, b0, (short)0, acc10, false, false);
            acc11 = __builtin_amdgcn_wmma_f32_16x16x4_f32(false, a1, false, b1, (short)0, acc11, false, false);
        }
        __syncthreads();
    }

    /* epilogue: C layout per CDNA5 ISA (vgpr p: M=p or 8+p; lanes: N) */
    const int gn = n0 + nw + lm;
    const int gm = m0 + mw + lh * 8;
    auto st = [&](int m, int n, float v) {
        if (bias) v += bias[(long)m * sbi_r + (long)n * sbi_c];
        if (sub)  v -= sub[(long)m * sc_r + n];
        Cp[(long)m * sc_r + n] = v;
    };
#pragma unroll
    for (int p = 0; p < 8; ++p) {
        st(gm + p,      gn,      acc00[p]);
        st(gm + p,      gn + 16, acc01[p]);
        st(gm + 16 + p, gn,      acc10[p]);
        st(gm + 16 + p, gn + 16, acc11[p]);
    }
}

/* ------------------------------------------------------------------ */
__global__ void transpose_2304(const float* __restrict__ src, float* __restrict__ dst)
{
    __shared__ float t[32][33];
    const int bx = blockIdx.x * 32, by = blockIdx.y * 32;
    const int tx = threadIdx.x, ty = threadIdx.y;   /* block (32,8) */
#pragma unroll
    for (int i = 0; i < 32; i += 8)
        t[ty + i][tx] = src[(long)(by + ty + i) * 2304 + bx + tx];
    __syncthreads();
#pragma unroll
    for (int i = 0; i < 32; i += 8)
        dst[(long)(bx + ty + i) * 2304 + by + tx] = t[tx][ty + i];
}

__global__ void fill_zero(float* __restrict__ p, long n)
{
    long i = (long)blockIdx.x * 256 + threadIdx.x;
    if (i < n) p[i] = 0.0f;
}

__global__ void elem_sq(const float* __restrict__ a, float* __restrict__ o, long n)
{
    long i = (long)blockIdx.x * 256 + threadIdx.x;
    if (i < n) { float u = a[i]; o[i] = u * u; }
}

__global__ void std_fin(const float* __restrict__ mean, float* __restrict__ msq, long n)
{
    long i = (long)blockIdx.x * 256 + threadIdx.x;
    if (i < n) {
        float m = mean[i];
        float d = msq[i] - m * m;
        msq[i] = sqrtf(fmaxf(d, 0.0f));
    }
}

__global__ void softmax_rows(float* __restrict__ X, int n)
{
    __shared__ float red[256];
    float* p = X + (long)blockIdx.x * n;
    const int t = threadIdx.x;
    float m = -3.402823e38f;
    for (int i = t; i < n; i += 256) m = fmaxf(m, p[i]);
    red[t] = m; __syncthreads();
    for (int s = 128; s > 0; s >>= 1) { if (t < s) red[t] = fmaxf(red[t], red[t + s]); __syncthreads(); }
    m = red[0]; __syncthreads();
    float sum = 0.0f;
    for (int i = t; i < n; i += 256) { float e = expf(p[i] - m); p[i] = e; sum += e; }
    red[t] = sum; __syncthreads();
    for (int s = 128; s > 0; s >>= 1) { if (t < s) red[t] += red[t + s]; __syncthreads(); }
    float inv = 1.0f / red[0];
    for (int i = t; i < n; i += 256) p[i] *= inv;
}

__global__ void init_v(float* __restrict__ v)
{
    int i = blockIdx.x * 256 + threadIdx.x;
    if (i < 2304) v[i] = 1.0f / 48.0f;   /* 1/sqrt(2304) */
}

__global__ void matvec2304(const float* __restrict__ W, const float* __restrict__ v,
                           float* __restrict__ w)
{
    const int row  = blockIdx.x * 8 + (threadIdx.x >> 5);
    const int lane = threadIdx.x & 31;
    const float* r = W + (long)row * 2304;
    float s = 0.0f;
    for (int i = lane; i < 2304; i += 32) s += r[i] * v[i];
    for (int o = 16; o > 0; o >>= 1) s += __shfl_down(s, o, 32);
    if (lane == 0) w[row] = s;
}

__global__ void v_normalize(const float* __restrict__ w, float* __restrict__ v)
{
    __shared__ float red[256];
    const int t = threadIdx.x;
    float s = 0.0f;
    for (int i = t; i < 2304; i += 256) { float u = w[i]; s += u * u; }
    red[t] = s; __syncthreads();
    for (int k = 128; k > 0; k >>= 1) { if (t < k) red[t] += red[t + k]; __syncthreads(); }
    float inv = 1.0f / sqrtf(red[0]);
    for (int i = t; i < 2304; i += 256) v[i] = w[i] * inv;
}

/* L = v.(WtW v);  scalars: [0]=t [1]=done [2]=lr [3]=lam [4]=beta [5]=tn [6]=diff */
__global__ void rayleigh_init(const float* __restrict__ v, const float* __restrict__ w,
                              float* __restrict__ sc)
{
    __shared__ float red[256];
    const int t = threadIdx.x;
    float s = 0.0f;
    for (int i = t; i < 2304; i += 256) s += v[i] * w[i];
    red[t] = s; __syncthreads();
    for (int k = 128; k > 0; k >>= 1) { if (t < k) red[t] += red[t + k]; __syncthreads(); }
    if (t == 0) {
        float lr = 1.0f / red[0];
        sc[0] = 1.0f; sc[1] = 0.0f; sc[2] = lr; sc[3] = 0.05f * lr;
        sc[4] = 0.0f; sc[5] = 0.0f; sc[6] = 0.0f;
    }
}

__global__ void groupnorm_combine(const float* __restrict__ contc,
                                  const float* __restrict__ stdc,
                                  const float* __restrict__ meanc,
                                  float* __restrict__ x)
{
    const long base = (long)blockIdx.x * 2304 + threadIdx.x * 9;
    float v[9], s = 0.0f, ss = 0.0f;
#pragma unroll
    for (int j = 0; j < 9; ++j) { float u = contc[base + j]; v[j] = u; s += u; ss += u * u; }
    float mu  = s * (1.0f / 9.0f);
    float var = (ss - 9.0f * mu * mu) * (1.0f / 8.0f);   /* unbiased */
    float rs  = rsqrtf(var + 1e-5f);
#pragma unroll
    for (int j = 0; j < 9; ++j)
        x[base + j] = (v[j] - mu) * rs * stdc[base + j] + meanc[base + j];
}

__global__ void fista_begin(float* __restrict__ sc)
{
    if (threadIdx.x == 0) {
        float t  = sc[0];
        float tn = 0.5f * (1.0f + sqrtf(1.0f + 4.0f * t * t));
        sc[5] = tn;
        sc[4] = (t - 1.0f) / tn;
        sc[6] = 0.0f;
    }
}

__global__ void fista_update(const float* __restrict__ grad, float* __restrict__ z,
                             float* __restrict__ y, float* __restrict__ sc)
{
    __shared__ float red[256];
    const long i = (long)blockIdx.x * 256 + threadIdx.x;
    const float lr = sc[2], lam = sc[3], beta = sc[4];
    const bool done = sc[1] != 0.0f;
    float ze = z[i], ye = y[i], g = grad[i];
    float u  = ye - lr * g;
    float au = fabsf(u) - lam;
    float zn = (au > 0.0f) ? copysignf(au, u) : 0.0f;   /* softshrink */
    float d  = fabsf(ze - zn);
    if (!done) { z[i] = zn; y[i] = zn + beta * (zn - ze); }
    red[threadIdx.x] = d; __syncthreads();
    for (int k = 128; k > 0; k >>= 1) { if (threadIdx.x < k) red[threadIdx.x] += red[threadIdx.x + k]; __syncthreads(); }
    if (threadIdx.x == 0) atomicAdd(&sc[6], red[0]);
}

__global__ void fista_end(float* __restrict__ sc)
{
    if (threadIdx.x == 0) {
        if (sc[1] == 0.0f) sc[0] = sc[5];       /* t2 = done ? t : tn */
        if (sc[6] <= TOLc) sc[1] = 1.0f;        /* done |= diff<=tol  */
    }
}

/* ------------------------------------------------------------------ */
static void gemm(hipStream_t st,
                 const float* A, long sar, long sac, long sab,
                 const float* B, long sbr, long sbc, long sbb,
                 float* C, long scr, long scb,
                 int Mr, int N, int K, int batches,
                 const float* bias, long sbir, long sbic,
                 const float* sub)
{
    dim3 grid(N / 128, Mr / 64, batches);
    wmma_gemm_f32<<<grid, 256, 0, st>>>(A, sar, sac, sab, B, sbr, sbc, sbb,
                                        C, scr, scb, bias, sbir, sbic, sub, K);
}

extern "C" void kernel_launch(void* const* d_in, const int* in_sizes, int n_in,
                              void* d_out, int out_size, void* d_ws, size_t ws_size,
                              hipStream_t stream)
{
    const float* content = (const float*)d_in[0];
    const float* style   = (const float*)d_in[1];
    const float* ckey    = (const float*)d_in[2];
    const float* skey    = (const float*)d_in[3];
    const float* f_w = (const float*)d_in[4];
    const float* f_b = (const float*)d_in[5];
    const float* g_w = (const float*)d_in[6];
    const float* g_b = (const float*)d_in[7];
    const float* h_w = (const float*)d_in[8];
    const float* h_b = (const float*)d_in[9];
    const float* Ain = (const float*)d_in[10];
    float* z  = (float*)d_out;
    float* ws = (float*)d_ws;

    /* workspace layout (floats), phase-overlapped */
    const long WS_NEED = 8192 + ATc + 2 * TOTc + (20054016L + 2 * TOTc);
    if ((long)(ws_size / 4) < WS_NEED) return;

    float* sc  = ws;
    float* v   = ws + 32;
    float* w   = ws + 32 + 2304;
    float* ATm = ws + 8192;            /* A^T [HW x M]           */
    float* X   = ATm + ATc;            /* FISTA target [B*C x M] */
    float* Y   = X + TOTc;             /* FISTA y      [B*C x M] */
    float* R   = Y + TOTc;             /* phase-shared region    */
    /* phase 1 */ float* WtW = R;
    /* phase 2 */ float* Fq = R;                 float* G   = R + TOTc;
                  float* Hm = R + 2 * TOTc;      float* Hsq = R + 3 * TOTc;
                  float* S  = R + 4 * TOTc;                      /* B*HW*HW */
                  float* Mean = S + (long)Bc * HWc * HWc;
                  float* Msq  = Mean + TOTc;                     /* -> std  */
                  float* MeanC = R;  float* StdC = R + TOTc;  float* ContC = R + 2 * TOTc;
    /* phase 3 */ float* Resid = R;    float* Grad = R + TOTc;

    const long bFeat = (long)Cc * HWc;   /* per-batch feature map */
    const long bQC   = (long)HWc * Cc;
    const long bS    = (long)HWc * HWc;

    /* ---------------- Phase 1: A^T, WtW, power iteration ---------- */
    transpose_2304<<<dim3(72, 72), dim3(32, 8), 0, stream>>>(Ain, ATm);
    gemm(stream, ATm, 2304, 1, 0, Ain, 2304, 1, 0, WtW, 2304, 0,
         2304, 2304, 2304, 1, nullptr, 0, 0, nullptr);
    init_v<<<9, 256, 0, stream>>>(v);
    for (int it = 0; it < 100; ++it) {
        matvec2304<<<288, 256, 0, stream>>>(WtW, v, w);
        v_normalize<<<1, 256, 0, stream>>>(w, v);
    }
    matvec2304<<<288, 256, 0, stream>>>(WtW, v, w);
    rayleigh_init<<<1, 256, 0, stream>>>(v, w, sc);

    /* ---------------- Phase 2: attention + projections ------------ */
    /* Fq[b] (HWxC) = ckey^T @ f_w^T + f_b (per-col bias)             */
    gemm(stream, ckey, 1, HWc, bFeat, f_w, 1, Cc, 0, Fq, Cc, bQC,
         HWc, Cc, Cc, Bc, f_b, 0, 1, nullptr);
    /* G[b] (CxHW) = g_w @ skey + g_b (per-row bias)                  */
    gemm(stream, g_w, Cc, 1, 0, skey, HWc, 1, bFeat, G, HWc, bFeat,
         Cc, HWc, Cc, Bc, g_b, 1, 0, nullptr);
    /* Hm[b] (HWxC) = style^T @ h_w^T + h_b                           */
    gemm(stream, style, 1, HWc, bFeat, h_w, 1, Cc, 0, Hm, Cc, bQC,
         HWc, Cc, Cc, Bc, h_b, 0, 1, nullptr);
    elem_sq<<<(int)(TOTc / 256), 256, 0, stream>>>(Hm, Hsq, TOTc);
    /* S[b] = Fq @ G ; softmax rows                                   */
    gemm(stream, Fq, Cc, 1, bQC, G, HWc, 1, bFeat, S, HWc, bS,
         HWc, HWc, Cc, Bc, nullptr, 0, 0, nullptr);
    softmax_rows<<<Bc * HWc, 256, 0, stream>>>(S, HWc);
    /* mean = S @ Hm ; msq = S @ Hsq ; std = sqrt(relu(msq-mean^2))   */
    gemm(stream, S, HWc, 1, bS, Hm, Cc, 1, bQC, Mean, Cc, bQC,
         HWc, Cc, HWc, Bc, nullptr, 0, 0, nullptr);
    gemm(stream, S, HWc, 1, bS, Hsq, Cc, 1, bQC, Msq, Cc, bQC,
         HWc, Cc, HWc, Bc, nullptr, 0, 0, nullptr);
    std_fin<<<(int)(TOTc / 256), 256, 0, stream>>>(Mean, Msq, TOTc);
    /* projections (C x M) = X^T(CxHW) @ ATm(HWxM)                    */
    gemm(stream, Mean, 1, Cc, bQC, ATm, Mc, 1, 0, MeanC, Mc, (long)Cc * Mc,
         Cc, Mc, HWc, Bc, nullptr, 0, 0, nullptr);
    gemm(stream, Msq, 1, Cc, bQC, ATm, Mc, 1, 0, StdC, Mc, (long)Cc * Mc,
         Cc, Mc, HWc, Bc, nullptr, 0, 0, nullptr);
    gemm(stream, content, HWc, 1, bFeat, ATm, Mc, 1, 0, ContC, Mc, (long)Cc * Mc,
         Cc, Mc, HWc, Bc, nullptr, 0, 0, nullptr);
    groupnorm_combine<<<Bc * Cc, 256, 0, stream>>>(ContC, StdC, MeanC, X);

    /* ---------------- Phase 3: FISTA (z lives in d_out) ----------- */
    fill_zero<<<(int)(TOTc / 256), 256, 0, stream>>>(z, TOTc);
    fill_zero<<<(int)(TOTc / 256), 256, 0, stream>>>(Y, TOTc);
    for (int it = 0; it < 43; ++it) {
        fista_begin<<<1, 32, 0, stream>>>(sc);
        /* resid = Y @ A^T - X   [1024 x 2304]                        */
        gemm(stream, Y, 2304, 1, 0, ATm, 2304, 1, 0, Resid, 2304, 0,
             Bc * Cc, Mc, HWc, 1, nullptr, 0, 0, X);
        /* grad = resid @ A      [1024 x 2304]                        */
        gemm(stream, Resid, 2304, 1, 0, Ain, 2304, 1, 0, Grad, 2304, 0,
             Bc * Cc, HWc, Mc, 1, nullptr, 0, 0, nullptr);
        fista_update<<<(int)(TOTc / 256), 256, 0, stream>>>(Grad, z, Y, sc);
        fista_end<<<1, 32, 0, stream>>>(sc);
    }
}